// SpatioTemporalGNN_17274358464909
// MI455X (gfx1250) — compile-verified
//
#include <hip/hip_runtime.h>

typedef float v2f __attribute__((ext_vector_type(2)));
typedef float v8f __attribute__((ext_vector_type(8)));

#define HID 64

// ---------------------------------------------------------------------------
// init: deg = 1.0 (self-loop weight), agg = 0, out = 0
// ---------------------------------------------------------------------------
__global__ void gcn_init_kernel(float* __restrict__ deg, float* __restrict__ agg,
                                float* __restrict__ out, int N) {
    int tid = blockIdx.x * blockDim.x + threadIdx.x;
    int total = N * HID;
    if (tid < N) deg[tid] = 1.0f;
    if (tid < total) { agg[tid] = 0.0f; out[tid] = 0.0f; }
}

// ---------------------------------------------------------------------------
// weighted in-degree accumulation over edge destinations
// ---------------------------------------------------------------------------
__global__ void gcn_degree_kernel(const int* __restrict__ dst, const float* __restrict__ w,
                                  float* __restrict__ deg, int E) {
    int e = blockIdx.x * blockDim.x + threadIdx.x;
    if (e < E) atomicAdd(&deg[dst[e]], w[e]);
}

// ---------------------------------------------------------------------------
// dinv = deg > 0 ? rsqrt(deg) : 0
// ---------------------------------------------------------------------------
__global__ void gcn_dinv_kernel(const float* __restrict__ deg, float* __restrict__ dinv, int N) {
    int i = blockIdx.x * blockDim.x + threadIdx.x;
    if (i < N) {
        float d = deg[i];
        dinv[i] = (d > 0.0f) ? rsqrtf(d) : 0.0f;
    }
}

// ---------------------------------------------------------------------------
// H = X @ W  via V_WMMA_F32_16X16X4_F32 (fp32 in/out, full precision)
// Block: 256 threads = 8 waves; covers 32 rows x 64 cols of output.
// Requires M % 32 == 0 (100000 = 3125 * 32). No divergence -> EXEC all ones.
// ---------------------------------------------------------------------------
__global__ __launch_bounds__(256) void gcn_gemm_wmma_kernel(
    const float* __restrict__ X, const float* __restrict__ W,
    float* __restrict__ Hout, int M) {
    __shared__ float Xs[32 * 65];   // padded stride 65 -> conflict-free A reads
    __shared__ float Ws[64 * 65];

    const int tid  = threadIdx.x;
    const int wave = tid >> 5;
    const int lane = tid & 31;
    const int m0   = blockIdx.x * 32;

    // Stage 32 rows of X (2048 floats) as float4, scatter into padded LDS.
    #pragma unroll
    for (int i = tid; i < 512; i += 256) {
        int r  = i >> 4;
        int c4 = (i & 15) << 2;
        const float4 v = *(const float4*)(X + (size_t)(m0 + r) * HID + c4);
        float* p = &Xs[r * 65 + c4];
        p[0] = v.x; p[1] = v.y; p[2] = v.z; p[3] = v.w;
    }
    // Stage full 64x64 W (4096 floats).
    #pragma unroll
    for (int i = tid; i < 1024; i += 256) {
        int r  = i >> 4;
        int c4 = (i & 15) << 2;
        const float4 v = *(const float4*)(W + r * HID + c4);
        float* p = &Ws[r * 65 + c4];
        p[0] = v.x; p[1] = v.y; p[2] = v.z; p[3] = v.w;
    }
    __syncthreads();

    const int mt = (wave >> 2) << 4;   // 0 or 16 (row tile within block)
    const int nt = (wave & 3) << 4;    // 0,16,32,48 (col tile)
    const int lr = lane & 15;
    const int hi = lane >> 4;          // 0: lanes 0-15, 1: lanes 16-31

    v8f c = {};
    #pragma unroll
    for (int k0 = 0; k0 < HID; k0 += 4) {
        const int ka = k0 + hi * 2;
        v2f a, b;
        // A 16x4: lane<16 holds (K=k0,k0+1) of row M=lr; lane>=16 holds (k0+2,k0+3)
        a.x = Xs[(mt + lr) * 65 + ka];
        a.y = Xs[(mt + lr) * 65 + ka + 1];
        // B 4x16 mirrored: VGPR0 = rows K=ka across N lanes, VGPR1 = K=ka+1
        b.x = Ws[ka * 65 + nt + lr];
        b.y = Ws[(ka + 1) * 65 + nt + lr];
        c = __builtin_amdgcn_wmma_f32_16x16x4_f32(
                /*neg_a=*/false, a, /*neg_b=*/false, b,
                /*c_mod=*/(short)0, c, /*reuse_a=*/false, /*reuse_b=*/false);
    }

    // C/D 16x16 f32 layout: VGPR r -> M = r (lanes 0-15) / r+8 (lanes 16-31), N = lane&15
    const size_t rowBase = (size_t)(m0 + mt + hi * 8);
    #pragma unroll
    for (int r = 0; r < 8; ++r) {
        Hout[(rowBase + r) * HID + nt + lr] = c[r];
    }
}

// ---------------------------------------------------------------------------
// Edge propagation: agg[dst] += h[src] * (dinv[src]*w*dinv[dst])
// 16 lanes per edge, float4 per lane. Working set is L2-resident (192MB L2).
// ---------------------------------------------------------------------------
__global__ void gcn_propagate_kernel(const float* __restrict__ h, const float* __restrict__ dinv,
                                     const int* __restrict__ src, const int* __restrict__ dst,
                                     const float* __restrict__ w, float* __restrict__ agg, int E) {
    long long tid = (long long)blockIdx.x * blockDim.x + threadIdx.x;
    int e = (int)(tid >> 4);
    if (e >= E) return;
    int f = (int)(tid & 15) << 2;
    int s = src[e];
    int d = dst[e];
    float norm = dinv[s] * w[e] * dinv[d];
    const float4 hv = *(const float4*)(h + (size_t)s * HID + f);
    float* o = agg + (size_t)d * HID + f;
    atomicAdd(o + 0, hv.x * norm);
    atomicAdd(o + 1, hv.y * norm);
    atomicAdd(o + 2, hv.z * norm);
    atomicAdd(o + 3, hv.w * norm);
}

// ---------------------------------------------------------------------------
// agg = relu(agg + h*dinv^2 + b)   (self-loop term + bias + ReLU, in place)
// ---------------------------------------------------------------------------
__global__ void gcn_relu_self_bias_kernel(float* __restrict__ agg, const float* __restrict__ h,
                                          const float* __restrict__ dinv, const float* __restrict__ b,
                                          int N) {
    int tid = blockIdx.x * blockDim.x + threadIdx.x;
    if (tid >= N * HID) return;
    int i = tid >> 6;
    int f = tid & 63;
    float di = dinv[i];
    float v = agg[tid] + h[tid] * di * di + b[f];
    agg[tid] = (v > 0.0f) ? v : 0.0f;
}

// ---------------------------------------------------------------------------
// out += h*dinv^2 + b   (final self-loop term + bias)
// ---------------------------------------------------------------------------
__global__ void gcn_final_self_bias_kernel(float* __restrict__ out, const float* __restrict__ h,
                                           const float* __restrict__ dinv, const float* __restrict__ b,
                                           int N) {
    int tid = blockIdx.x * blockDim.x + threadIdx.x;
    if (tid >= N * HID) return;
    int i = tid >> 6;
    int f = tid & 63;
    float di = dinv[i];
    out[tid] = out[tid] + h[tid] * di * di + b[f];
}

// ---------------------------------------------------------------------------
extern "C" void kernel_launch(void* const* d_in, const int* in_sizes, int n_in,
                              void* d_out, int out_size, void* d_ws, size_t ws_size,
                              hipStream_t stream) {
    const float* x  = (const float*)d_in[0];
    const int*   ei = (const int*)d_in[1];
    const float* ew = (const float*)d_in[2];
    const float* W1 = (const float*)d_in[3];
    const float* b1 = (const float*)d_in[4];
    const float* W2 = (const float*)d_in[5];
    const float* b2 = (const float*)d_in[6];

    const int N = in_sizes[0] / HID;     // 100000
    const int E = in_sizes[2];           // 1200000
    const int* src = ei;                 // edge_index[0]
    const int* dst = ei + E;             // edge_index[1]

    // Workspace layout (floats): deg[N] | dinv[N] | h[N*64] | agg[N*64]  (~52 MB)
    float* deg  = (float*)d_ws;
    float* dinv = deg + N;
    float* h    = dinv + N;
    float* agg  = h + (size_t)N * HID;
    float* out  = (float*)d_out;

    const int NF = N * HID;
    const int blk = 256;
    const int gNF = (NF + blk - 1) / blk;
    const int gE  = (E + blk - 1) / blk;
    const int gN  = (N + blk - 1) / blk;
    const long long ef = (long long)E * 16;
    const int gEF = (int)((ef + blk - 1) / blk);
    const int gGemm = N / 32;            // 100000 % 32 == 0

    // Normalization
    gcn_init_kernel<<<gNF, blk, 0, stream>>>(deg, agg, out, N);
    gcn_degree_kernel<<<gE, blk, 0, stream>>>(dst, ew, deg, E);
    gcn_dinv_kernel<<<gN, blk, 0, stream>>>(deg, dinv, N);

    // Layer 1
    gcn_gemm_wmma_kernel<<<gGemm, blk, 0, stream>>>(x, W1, h, N);
    gcn_propagate_kernel<<<gEF, blk, 0, stream>>>(h, dinv, src, dst, ew, agg, E);
    gcn_relu_self_bias_kernel<<<gNF, blk, 0, stream>>>(agg, h, dinv, b1, N);

    // Layer 2 (reuse h as h2; agg now holds relu'd layer-1 output)
    gcn_gemm_wmma_kernel<<<gGemm, blk, 0, stream>>>(agg, W2, h, N);
    gcn_propagate_kernel<<<gEF, blk, 0, stream>>>(h, dinv, src, dst, ew, out, E);
    gcn_final_self_bias_kernel<<<gNF, blk, 0, stream>>>(out, h, dinv, b2, N);
}